// TropicalNewton_5531917877920
// MI455X (gfx1250) — compile-verified
//
#include <hip/hip_runtime.h>
#include <float.h>

#define NDIM 8192
#define REGC 1e-6f

typedef int v4i __attribute__((ext_vector_type(4)));

// ---------------- CDNA5 async-to-LDS support (probe-guarded) ----------------
#if defined(__AMDGCN__) && __has_builtin(__builtin_amdgcn_global_load_async_to_lds_b128)
#define USE_ASYNC_LDS 1
typedef __attribute__((address_space(1))) v4i gv4i_t;   // global 16B vector
typedef __attribute__((address_space(3))) v4i lv4i_t;   // LDS 16B vector
#else
#define USE_ASYNC_LDS 0
#endif

__device__ __forceinline__ void wait_async0() {
#if defined(__AMDGCN__)
#if __has_builtin(__builtin_amdgcn_s_wait_asynccnt)
  __builtin_amdgcn_s_wait_asynccnt(0);
#else
  asm volatile("s_wait_asynccnt 0" ::: "memory");
#endif
#endif
}

// ------------- monotone float <-> uint encoding for atomic min -------------
__device__ __forceinline__ unsigned encf(float f) {
  unsigned u = __float_as_uint(f);
  return (u & 0x80000000u) ? ~u : (u | 0x80000000u);
}
__device__ __forceinline__ float decf(unsigned u) {
  return __uint_as_float((u & 0x80000000u) ? (u & 0x7FFFFFFFu) : ~u);
}

// ---------------------------------------------------------------------------
__global__ void k_init_bits(unsigned* __restrict__ xbits) {
  int j = blockIdx.x * blockDim.x + threadIdx.x;
  if (j < NDIM) xbits[j] = 0xFFFFFFFFu;  // encoded +inf (identity for min)
}

// Column-wise min of (g[i] - H[i,j] - (i==j)*reg), accumulated via atomicMin
// on encoded bits. Grid: (8 column groups of 1024, 128 row chunks of 64).
#define CM_ROWS 64
__global__ void k_colmin(const float* __restrict__ H, const float* __restrict__ g,
                         unsigned* __restrict__ xbits) {
  __shared__ float gs[CM_ROWS];
  const int t  = threadIdx.x;
  const int j0 = blockIdx.x * 1024 + t * 4;   // this thread's 4 columns
  const int i0 = blockIdx.y * CM_ROWS;        // row chunk base
  if (t < CM_ROWS) gs[t] = g[i0 + t];
  __syncthreads();

  float4 m = make_float4(FLT_MAX, FLT_MAX, FLT_MAX, FLT_MAX);
  const float* p = H + (size_t)i0 * NDIM + j0;
  for (int r = 0; r < CM_ROWS; ++r) {
    const float4 h = *(const float4*)p;
    p += NDIM;
    const float gi = gs[r];
    const int i = i0 + r;
    m.x = fminf(m.x, gi - h.x - ((j0 + 0) == i ? REGC : 0.0f));
    m.y = fminf(m.y, gi - h.y - ((j0 + 1) == i ? REGC : 0.0f));
    m.z = fminf(m.z, gi - h.z - ((j0 + 2) == i ? REGC : 0.0f));
    m.w = fminf(m.w, gi - h.w - ((j0 + 3) == i ? REGC : 0.0f));
  }
  atomicMin(&xbits[j0 + 0], encf(m.x));
  atomicMin(&xbits[j0 + 1], encf(m.y));
  atomicMin(&xbits[j0 + 2], encf(m.z));
  atomicMin(&xbits[j0 + 3], encf(m.w));
}

__global__ void k_decode(float* __restrict__ x) {
  int j = blockIdx.x * blockDim.x + threadIdx.x;
  if (j < NDIM) {
    unsigned u = ((const unsigned*)x)[j];
    x[j] = decf(u);
  }
}

// Row-wise max of (H[i,j] + x[j] + (i==j)*reg). One block per row.
// x[] (32 KB) is staged into LDS via CDNA5 async global->LDS b128 copies.
__global__ void k_rowmax(const float* __restrict__ H, const float* __restrict__ x,
                         float* __restrict__ bhat) {
  __shared__ float xs[NDIM];   // 32 KB
  __shared__ float red[256];
  const int i = blockIdx.x;
  const int t = threadIdx.x;

#if USE_ASYNC_LDS
  // 2048 float4s staged by 256 threads: 8 async b128 copies per lane.
  #pragma unroll
  for (int k = 0; k < 8; ++k) {
    const int e = (t + k * 256) * 4;  // element index (16B aligned)
    __builtin_amdgcn_global_load_async_to_lds_b128(
        (gv4i_t*)(x + e), (lv4i_t*)(&xs[e]), 0, 0);
  }
  wait_async0();
  __syncthreads();
#else
  #pragma unroll
  for (int k = 0; k < 8; ++k) {
    const int e = (t + k * 256) * 4;
    *(float4*)(&xs[e]) = *(const float4*)(x + e);
  }
  __syncthreads();
#endif

  const float* row = H + (size_t)i * NDIM;
  float m = -FLT_MAX;
  #pragma unroll
  for (int k = 0; k < 8; ++k) {
    const int e = (t + k * 256) * 4;
    const float4 h  = *(const float4*)(row + e);
    const float4 xv = *(const float4*)(&xs[e]);
    m = fmaxf(m, h.x + xv.x + ((e + 0) == i ? REGC : 0.0f));
    m = fmaxf(m, h.y + xv.y + ((e + 1) == i ? REGC : 0.0f));
    m = fmaxf(m, h.z + xv.z + ((e + 2) == i ? REGC : 0.0f));
    m = fmaxf(m, h.w + xv.w + ((e + 3) == i ? REGC : 0.0f));
  }
  red[t] = m;
  __syncthreads();
  for (int s = 128; s > 0; s >>= 1) {
    if (t < s) red[t] = fmaxf(red[t], red[t + s]);
    __syncthreads();
  }
  if (t == 0) bhat[i] = red[0];
}

// ---------------------------------------------------------------------------
extern "C" void kernel_launch(void* const* d_in, const int* in_sizes, int n_in,
                              void* d_out, int out_size, void* d_ws, size_t ws_size,
                              hipStream_t stream) {
  (void)in_sizes; (void)n_in; (void)out_size; (void)d_ws; (void)ws_size;
  const float* H = (const float*)d_in[0];
  const float* g = (const float*)d_in[1];
  float* out = (float*)d_out;            // out[0..8191] = x, out[8192..16383] = b_hat
  unsigned* xbits = (unsigned*)out;

  k_init_bits<<<NDIM / 256, 256, 0, stream>>>(xbits);
  dim3 gcm(NDIM / 1024, NDIM / CM_ROWS);           // (8, 128)
  k_colmin<<<gcm, 256, 0, stream>>>(H, g, xbits);
  k_decode<<<NDIM / 256, 256, 0, stream>>>(out);
  k_rowmax<<<NDIM, 256, 0, stream>>>(H, out, out + NDIM);
}